// ModuleCorrelation_1185410974101
// MI455X (gfx1250) — compile-verified
//
#include <hip/hip_runtime.h>
#include <hip/hip_bf16.h>

// PWC-Net correlation: out[b, (dy+4)*9+(dx+4), h, w] =
//   mean_c( first[b,c,h,w] * second[b,c,h+dy,w+dx] ), zero-padded.
// Mapped onto V_WMMA_F32_16X16X4_F32: banded GEMM over 16-pixel row tiles.
// K-loop fully unrolled so all global loads use immediate 24-bit offsets
// (max 7.68MB < 8.38MB limit) and the prefetch guard constant-folds.

typedef float v2f __attribute__((ext_vector_type(2)));
typedef float v8f __attribute__((ext_vector_type(8)));

#define BATCH   8
#define CHANS   128
#define HEIGHT  96
#define WIDTH   160
#define DISP    4
#define GRIDW   9              // 2*DISP+1
#define HW      (HEIGHT * WIDTH)

__global__ __launch_bounds__(32)
void corr_wmma_kernel(const float* __restrict__ first,
                      const float* __restrict__ second,
                      float* __restrict__ out)
{
    const int lane = threadIdx.x;      // 0..31 (wave32)
    const int m    = lane & 15;        // tile row (A) / tile column (B)
    const int half = lane >> 4;        // 0 or 1
    const int kk   = half * 2;         // K sub-offset handled by this half-wave

    const int w0    = blockIdx.x * 16; // first output column of this tile
    const int h     = blockIdx.y;
    const int dyIdx = blockIdx.z % GRIDW;   // 0..8  (dy = dyIdx - 4)
    const int b     = blockIdx.z / GRIDW;

    const int  hrow  = h + dyIdx - DISP;    // source row in `second`
    const bool rowOK = (hrow >= 0) && (hrow < HEIGHT);

    // G[i, j'] = sum_c F[c, w0+i] * S[c, hrow, w0-4+j']
    v8f acc_lo = {};   // columns j' = 0..15  -> w = w0-4 .. w0+11
    v8f acc_hi = {};   // columns j' = 8..23  -> w = w0+4 .. w0+19

    if (rowOK) {
        // A: lane holds F[c0+kk + {0,1}, h, w0+m]
        const float* pA = first + ((size_t)(b * CHANS + kk) * HEIGHT + h) * WIDTH
                                + (size_t)(w0 + m);

        // B columns for this lane. Edge masking is multiplicative through the
        // dot product: clamp the address in-range (unconditional, safe load)
        // and scale the value by a hoisted 0/1 mask -> no EXEC branches.
        const int   colLo = w0 - DISP + m;     // j' = m
        const int   colHi = colLo + 8;         // j' = m + 8  (>= 4 always)
        const float mLo = (colLo >= 0 && colLo < WIDTH) ? 1.0f : 0.0f;
        const float mHi = (colHi < WIDTH) ? 1.0f : 0.0f;
        int colLoC = colLo < 0 ? 0 : (colLo >= WIDTH ? WIDTH - 1 : colLo);
        int colHiC = colHi >= WIDTH ? WIDTH - 1 : colHi;

        const size_t sbase = ((size_t)(b * CHANS + kk) * HEIGHT + hrow) * WIDTH;
        const float* pBlo = second + sbase + (size_t)colLoC;
        const float* pBhi = second + sbase + (size_t)colHiC;

        #pragma unroll
        for (int c0 = 0; c0 < CHANS; c0 += 4) {
            const size_t off = (size_t)c0 * HW;   // compile-time after unroll
            v2f a, blo, bhi;
            a.x   = pA[off];
            a.y   = pA[off + HW];
            blo.x = pBlo[off]      * mLo;
            blo.y = pBlo[off + HW] * mLo;
            bhi.x = pBhi[off]      * mHi;
            bhi.y = pBhi[off + HW] * mHi;

            if (c0 + 32 < CHANS) {   // constant-folds per unrolled iteration
                __builtin_prefetch(pA + off + 32 * (size_t)HW, 0, 3);
            }

            // D = A x B + C   (16x16x4 fp32 WMMA)
            acc_lo = __builtin_amdgcn_wmma_f32_16x16x4_f32(
                         false, a, false, blo, (short)0, acc_lo, false, false);
            acc_hi = __builtin_amdgcn_wmma_f32_16x16x4_f32(
                         false, a, false, bhi, (short)0, acc_hi, false, false);
        }
    }

    // Spill both 16x16 f32 accumulator tiles to LDS for band extraction.
    // C/D layout: VGPR r, lanes 0-15 -> (M=r, N=lane); lanes 16-31 -> (M=r+8, N=lane-16)
    __shared__ float lds[2][16][16];
    #pragma unroll
    for (int r = 0; r < 8; ++r) {
        lds[0][r + 8 * half][m] = acc_lo[r];
        lds[1][r + 8 * half][m] = acc_hi[r];
    }
    __syncthreads();

    // 144 outputs per tile: pixel row i (output w = w0+i), horiz offset ox = dx+4.
    //   i < 8 : value = G_lo[i, i+ox]
    //   i >= 8: value = G_hi[i, i+ox-8]
    const float scale = 1.0f / (float)CHANS;
    for (int e = lane; e < 16 * GRIDW; e += 32) {
        const int i  = e / GRIDW;
        const int ox = e % GRIDW;
        const float v = (i < 8) ? lds[0][i][i + ox] : lds[1][i][i + ox - 8];
        const int ch = dyIdx * GRIDW + ox;   // (dy+4)*9 + (dx+4)
        out[(((size_t)b * (GRIDW * GRIDW) + ch) * HEIGHT + h) * WIDTH + (w0 + i)]
            = v * scale;
    }
}

extern "C" void kernel_launch(void* const* d_in, const int* in_sizes, int n_in,
                              void* d_out, int out_size, void* d_ws, size_t ws_size,
                              hipStream_t stream) {
    (void)in_sizes; (void)n_in; (void)out_size; (void)d_ws; (void)ws_size;
    const float* first  = (const float*)d_in[0];
    const float* second = (const float*)d_in[1];
    float*       out    = (float*)d_out;

    dim3 grid(WIDTH / 16, HEIGHT, BATCH * GRIDW);   // (10, 96, 72)
    dim3 block(32);                                  // one wave32 per block
    hipLaunchKernelGGL(corr_wmma_kernel, grid, block, 0, stream,
                       first, second, out);
}